// PairNormGCN_41987600285801
// MI455X (gfx1250) — compile-verified
//
#include <hip/hip_runtime.h>

#define D_H 128
#define PN_EPS 1e-6f

typedef float v2f __attribute__((ext_vector_type(2)));
typedef float v8f __attribute__((ext_vector_type(8)));

// -------------------------------------------------------------------------
// H[n, DOUT] = X[n, 128] @ W[128, DOUT] + bias      (fp32 WMMA 16x16x4)
//
// Block = 256 threads = 8 waves. Wave w computes the full 16-row x DOUT
// strip for M-tile (blockIdx.x*8 + w): the A fragment is loaded once per
// k-step and reused across all TN=DOUT/16 column tiles.
//
// W is staged into LDS TRANSPOSED in 64-row K-chunks: Wt[col][k], row
// stride 68 floats. A lane's B fragment (W[k+koff][col], W[k+koff+1][col])
// is then 8 contiguous aligned bytes -> one ds_load_b64 straight into an
// even VGPR pair (no repack movs), with all addressing folded into DS
// immediate offsets. 68 % 64 == 4, so lanes 0-15 hit bank pairs {4l,4l+1}
// and lanes 16-31 (K offset +2) hit {4l+2,4l+3}: all 64 banks exactly
// once -> conflict-free B reads.
// -------------------------------------------------------------------------
template <int DOUT>
__global__ void __launch_bounds__(256)
gemm_wmma_f32(const float* __restrict__ X, const float* __restrict__ W,
              const float* __restrict__ bias, float* __restrict__ H,
              int n_rows) {
    constexpr int TN  = DOUT / 16;   // column tiles per wave (8 or 4)
    constexpr int KC  = 64;          // K rows staged per chunk
    constexpr int WST = KC + 4;      // 68: padded transposed row stride
    __shared__ float Wt[DOUT * WST]; // 34.8 KB (DOUT=128) / 17.4 KB (DOUT=64)

    const int lane = threadIdx.x & 31;
    const int wave = threadIdx.x >> 5;
    const int half = lane >> 4;      // 0: K=k+0,k+1   1: K=k+2,k+3
    const int l15  = lane & 15;

    const int tm      = blockIdx.x * 8 + wave;
    const int tiles_m = (n_rows + 15) >> 4;
    const bool active = (tm < tiles_m);              // wave-uniform

    int row_a = tm * 16 + l15;
    if (row_a >= n_rows) row_a = n_rows - 1;         // clamp, EXEC stays full
    const float* __restrict__ ap = X + (size_t)row_a * D_H + half * 2;
    const float* __restrict__ bbase = &Wt[l15 * WST + half * 2];

    v8f acc[TN] = {};

    for (int k0 = 0; k0 < D_H; k0 += KC) {
        __syncthreads();                             // protect previous chunk
        const float* __restrict__ wg = W + (size_t)k0 * DOUT;
        for (int idx = threadIdx.x; idx < KC * DOUT; idx += 256) {
            int r = idx / DOUT;                      // K row within chunk
            int c = idx - r * DOUT;                  // output column
            Wt[c * WST + r] = wg[idx];               // transpose into LDS
        }
        __syncthreads();

        if (active) {
            __builtin_prefetch(ap + k0 + KC, 0, 1);  // next A chunk
            for (int kk = 0; kk < KC; kk += 4) {
                v2f a = *(const v2f*)(ap + k0 + kk);
#pragma unroll
                for (int t = 0; t < TN; ++t) {
                    // 8B-aligned contiguous pair -> ds_load_b64, immediate offset
                    v2f b = *(const v2f*)(bbase + t * (16 * WST) + kk);
                    acc[t] = __builtin_amdgcn_wmma_f32_16x16x4_f32(
                        /*neg_a=*/false, a, /*neg_b=*/false, b,
                        /*c_mod=*/(short)0, acc[t],
                        /*reuse_a=*/false, /*reuse_b=*/false);
                }
            }
        }
    }

    if (!active) return;
    const int row_base = tm * 16 + half * 8;         // C/D layout: half -> M+8
#pragma unroll
    for (int t = 0; t < TN; ++t) {
        const int col = t * 16 + l15;
        const float bb = bias[col];
#pragma unroll
        for (int r = 0; r < 8; ++r) {
            const int row = row_base + r;
            if (row < n_rows)
                H[(size_t)row * DOUT + col] = acc[t][r] + bb;
        }
    }
}

// -------------------------------------------------------------------------
// out[dst[e], :] += ew[e] * H[src[e], :]   — one 32-lane group per edge,
// each lane owns D/32 contiguous floats; hardware f32 atomics resolve in
// the 192 MB L2 (node matrix = 51 MB, fully resident).
// -------------------------------------------------------------------------
__global__ void spmm_scatter(const float* __restrict__ H,
                             const int* __restrict__ src,
                             const int* __restrict__ dst,
                             const float* __restrict__ ew,
                             float* __restrict__ out, int nE, int D) {
    const int lane = threadIdx.x & 31;
    const int e = (int)((blockIdx.x * (long long)blockDim.x + threadIdx.x) >> 5);
    if (e >= nE) return;
    const int s = src[e];
    const int d = dst[e];
    const float w = ew[e];
    const int per = D >> 5;                          // 4 (D=128) or 2 (D=64)
    const float* __restrict__ hp = H + (long long)s * D + lane * per;
    float* __restrict__ op = out + (long long)d * D + lane * per;
    for (int i = 0; i < per; ++i)
        unsafeAtomicAdd(&op[i], w * hp[i]);          // global_atomic_add_f32
}

__global__ void zero_f32(float* __restrict__ p, long long count) {
    long long i = blockIdx.x * (long long)blockDim.x + threadIdx.x;
    const long long stride = (long long)gridDim.x * blockDim.x;
    for (; i < count; i += stride) p[i] = 0.0f;
}

// -------------------------------------------------------------------------
// PairNorm reduction: red[c] = sum_i H[i,c] (c<128), red[128] = sum H^2.
// Centered sum-of-squares recovered later as  red[128] - N * sum(mean^2).
// -------------------------------------------------------------------------
__global__ void __launch_bounds__(128)
pn_reduce(const float* __restrict__ H, float* __restrict__ red, int n_rows) {
    __shared__ float ssq[D_H];
    const int c = threadIdx.x;                       // one thread per column
    float colsum = 0.0f, sumsq = 0.0f;
    for (int r = blockIdx.x; r < n_rows; r += gridDim.x) {
        float v = H[(long long)r * D_H + c];
        colsum += v;
        sumsq += v * v;
    }
    unsafeAtomicAdd(&red[c], colsum);
    ssq[c] = sumsq;
    __syncthreads();
    for (int s = 64; s > 0; s >>= 1) {
        if (c < s) ssq[c] += ssq[c + s];
        __syncthreads();
    }
    if (c == 0) unsafeAtomicAdd(&red[D_H], ssq[0]);
}

// red[c] -> column mean; red[129] -> 1/sqrt(eps + mean centered rownorm^2)
__global__ void __launch_bounds__(128)
pn_finalize(float* __restrict__ red, int n_rows) {
    __shared__ float sh[D_H];
    const int c = threadIdx.x;
    const float inv_n = 1.0f / (float)n_rows;
    const float mean = red[c] * inv_n;
    sh[c] = mean * mean;
    __syncthreads();
    for (int s = 64; s > 0; s >>= 1) {
        if (c < s) sh[c] += sh[c + s];
        __syncthreads();
    }
    if (c == 0) {
        float ssq_centered = red[D_H] - (float)n_rows * sh[0];
        red[D_H + 1] = 1.0f / sqrtf(PN_EPS + ssq_centered * inv_n);
    }
    red[c] = mean;
}

// xout = relu((conv - mean_col) * scale) + (xold ? xold : 0)
__global__ void pn_apply(const float* __restrict__ conv,
                         const float* __restrict__ xold,
                         float* __restrict__ xout,
                         const float* __restrict__ red, int n_rows) {
    const long long i = blockIdx.x * (long long)blockDim.x + threadIdx.x;
    if (i >= (long long)n_rows * D_H) return;
    const int c = (int)(i & (D_H - 1));
    float v = (conv[i] - red[c]) * red[D_H + 1];
    v = fmaxf(v, 0.0f);
    if (xold) v += xold[i];
    xout[i] = v;
}

// -------------------------------------------------------------------------
extern "C" void kernel_launch(void* const* d_in, const int* in_sizes, int n_in,
                              void* d_out, int out_size, void* d_ws, size_t ws_size,
                              hipStream_t stream) {
    const float* x    = (const float*)d_in[0];
    const int*   ei   = (const int*)d_in[1];
    const float* ew   = (const float*)d_in[2];
    const float* W0   = (const float*)d_in[3];
    const float* b0   = (const float*)d_in[4];
    const float* W1   = (const float*)d_in[5];
    const float* b1   = (const float*)d_in[6];
    const float* Wout = (const float*)d_in[7];
    const float* bout = (const float*)d_in[8];
    float* out = (float*)d_out;

    const int n = in_sizes[0] / D_H;                 // 100000
    const int e = in_sizes[2];                       // 1600000
    const int d_outc = out_size / n;                 // 64
    const int* srcI = ei;                            // edge_index[0,:]
    const int* dstI = ei + e;                        // edge_index[1,:]

    const size_t nd = (size_t)n * D_H;
    float* A   = (float*)d_ws;                       // lin out / x1 / lin_out
    float* B   = A + nd;                             // conv1 out / lin1 out
    float* C   = B + nd;                             // conv2 out / x2
    float* red = C + nd;                             // 256 floats of scratch

    const int gemm_blocks = (((n + 15) / 16) + 7) / 8;   // 8 M-tiles per block
    auto launch_gemm = [&](const float* X_, const float* W_, const float* bias_,
                           float* H_, int dout) {
        if (dout == 128)
            gemm_wmma_f32<128><<<gemm_blocks, 256, 0, stream>>>(X_, W_, bias_, H_, n);
        else
            gemm_wmma_f32<64><<<gemm_blocks, 256, 0, stream>>>(X_, W_, bias_, H_, n);
    };
    auto launch_scatter = [&](const float* H_, float* O_, int D) {
        long long thr = (long long)e * 32;
        int blocks = (int)((thr + 255) / 256);
        spmm_scatter<<<blocks, 256, 0, stream>>>(H_, srcI, dstI, ew, O_, e, D);
    };
    auto launch_zero = [&](float* p, long long cnt) {
        int blocks = (int)((cnt + 256 * 8 - 1) / (256 * 8));
        zero_f32<<<blocks, 256, 0, stream>>>(p, cnt);
    };
    auto launch_pn = [&](const float* conv, const float* xold, float* xout) {
        launch_zero(red, 256);
        pn_reduce<<<1024, 128, 0, stream>>>(conv, red, n);
        pn_finalize<<<1, 128, 0, stream>>>(red, n);
        long long cnt = (long long)n * D_H;
        pn_apply<<<(int)((cnt + 255) / 256), 256, 0, stream>>>(conv, xold, xout, red, n);
    };

    // Layer 0: x1 = relu(pairnorm(spmm(x @ W0 + b0)))
    launch_gemm(x, W0, b0, A, D_H);
    launch_zero(B, (long long)nd);
    launch_scatter(A, B, D_H);
    launch_pn(B, nullptr, A);                        // x1 -> A

    // Layer 1: x2 = relu(pairnorm(spmm(x1 @ W1 + b1))) + x1
    launch_gemm(A, W1, b1, B, D_H);
    launch_zero(C, (long long)nd);
    launch_scatter(B, C, D_H);
    launch_pn(C, A, C);                              // x2 -> C (in-place + residual)

    // Output layer: out = spmm(x2 @ Wout + bout)
    launch_gemm(C, Wout, bout, A, d_outc);
    launch_zero(out, (long long)n * d_outc);
    launch_scatter(A, out, d_outc);
}